// VQVAE_28089086115946
// MI455X (gfx1250) — compile-verified
//
#include <hip/hip_runtime.h>

typedef __attribute__((ext_vector_type(2))) float v2f;
typedef __attribute__((ext_vector_type(4))) float v4f;
typedef __attribute__((ext_vector_type(8))) float v8f;

#define NPOINTS 32768   // B*H*W = 8*64*64
#define KCODE   4096
#define HWSZ    4096    // H*W
#define CHW     16384   // C*H*W
#define LOSS0   131072
#define LOSS1   131073
#define IDXOFF  131074

// One wave handles 64 points (4 A-tiles of 16x4), sweeps all 4096 codebook
// entries in 256 B-tiles of 4x16 via V_WMMA_F32_16X16X4_F32.
// Score tile D = (-2x) * e + ||e||^2  (||x||^2 dropped: constant per row).
// The emb tile load is software-pipelined one iteration ahead so its L2
// latency overlaps the 4 WMMAs + argmin VALU of the previous tile.
__global__ __launch_bounds__(256)
void vq_argmin_kernel(const float* __restrict__ x, const float* __restrict__ emb,
                      float* __restrict__ out, float* __restrict__ partial)
{
    __shared__ int   s_idx[512];   // per-block winning codebook index per point
    __shared__ float s_red[8];

    const int tid   = threadIdx.x;
    const int wave  = tid >> 5;
    const int lane  = tid & 31;
    const int half  = lane >> 4;   // 0: K=0,1 / rows 0..7 ; 1: K=2,3 / rows 8..15
    const int mrow  = lane & 15;   // A row (M) and B column (N) for this lane
    const int gwave = blockIdx.x * 8 + wave;
    const int base  = gwave * 64;  // first point of this wave

    // ---- Load A tiles: lane holds point (base+mt*16+mrow), channels c0,c0+1,
    //      scaled by -2 so WMMA accumulates -2*x.e directly. x is NCHW. ----
    v2f a[4];
    const int c0 = half * 2;
#pragma unroll
    for (int mt = 0; mt < 4; ++mt) {
        int n  = base + mt * 16 + mrow;
        int b  = n >> 12;            // n / (H*W)
        int hw = n & (HWSZ - 1);
        const float* px = x + b * CHW + c0 * HWSZ + hw;
        a[mt].x = -2.0f * px[0];
        a[mt].y = -2.0f * px[HWSZ];
    }

    float bestv[4][8];
    int   bidx [4][8];
#pragma unroll
    for (int mt = 0; mt < 4; ++mt)
#pragma unroll
        for (int j = 0; j < 8; ++j) { bestv[mt][j] = 3.4e38f; bidx[mt][j] = 0; }

    // ---- Sweep codebook: 256 tiles of 16 entries, 4 WMMAs per tile.
    //      Double-buffered emb load: issue tile t+1 before consuming tile t. ----
    v4f e = *(const v4f*)(emb + mrow * 4);   // tile 0
#pragma unroll 2
    for (int t = 0; t < 256; ++t) {
        // prefetch next tile (wraps to tile 0 on last iter; branch-free)
        v4f en = *(const v4f*)(emb + ((((t + 1) & 255) * 16 + mrow) * 4));

        const int kidx = t * 16 + mrow;           // this lane's B column entry
        float nrm = e.x * e.x + e.y * e.y + e.z * e.z + e.w * e.w;
        // B 4x16 f32 layout: lanes 0-15 hold K=0 (v0), K=1 (v1);
        //                    lanes 16-31 hold K=2 (v0), K=3 (v1); N = lane%16.
        v2f bv;
        bv.x = half ? e.z : e.x;
        bv.y = half ? e.w : e.y;
        v8f c = {nrm, nrm, nrm, nrm, nrm, nrm, nrm, nrm};
#pragma unroll
        for (int mt = 0; mt < 4; ++mt) {
            v8f d = __builtin_amdgcn_wmma_f32_16x16x4_f32(
                false, a[mt], false, bv, (short)0, c, false, false);
            // D layout: VGPR j = row (j + half*8), column = mrow. Track per-lane
            // running min over this column's entries (branchless).
#pragma unroll
            for (int j = 0; j < 8; ++j) {
                float dj = d[j];
                bool lt  = dj < bestv[mt][j];
                bestv[mt][j] = lt ? dj   : bestv[mt][j];
                bidx [mt][j] = lt ? kidx : bidx[mt][j];
            }
        }
        e = en;
    }

    // ---- Cross-lane argmin per row: butterfly within each 16-lane half.
    //      Tie-break on smaller index (jnp.argmin takes first occurrence). ----
#pragma unroll
    for (int mt = 0; mt < 4; ++mt) {
#pragma unroll
        for (int j = 0; j < 8; ++j) {
            float v  = bestv[mt][j];
            int   ix = bidx [mt][j];
#pragma unroll
            for (int off = 8; off >= 1; off >>= 1) {
                float ov = __shfl_xor(v, off, 32);
                int   oi = __shfl_xor(ix, off, 32);
                if (ov < v || (ov == v && oi < ix)) { v = ov; ix = oi; }
            }
            if (mrow == j)   // one lane per half writes row (j + half*8)
                s_idx[wave * 64 + mt * 16 + half * 8 + j] = ix;
        }
    }
    __syncthreads();

    // ---- Phase 2: gather quant, write NCHW output + indices, local loss ----
    float lsum = 0.0f;
#pragma unroll
    for (int r = 0; r < 2; ++r) {
        int p  = tid + r * 256;
        int n  = blockIdx.x * 512 + p;
        int k  = s_idx[p];
        v4f ev = *(const v4f*)(emb + k * 4);
        int b  = n >> 12;
        int hw = n & (HWSZ - 1);
        const float* px = x   + b * CHW + hw;
        float*       po = out + b * CHW + hw;
        float d0 = ev.x - px[0 * HWSZ];
        float d1 = ev.y - px[1 * HWSZ];
        float d2 = ev.z - px[2 * HWSZ];
        float d3 = ev.w - px[3 * HWSZ];
        po[0 * HWSZ] = ev.x;  // quant_out == quant numerically (straight-through)
        po[1 * HWSZ] = ev.y;
        po[2 * HWSZ] = ev.z;
        po[3 * HWSZ] = ev.w;
        out[IDXOFF + n] = (float)k;
        lsum += d0 * d0 + d1 * d1 + d2 * d2 + d3 * d3;
    }

    // deterministic block reduction -> partial[blockIdx.x]
#pragma unroll
    for (int off = 16; off >= 1; off >>= 1) lsum += __shfl_xor(lsum, off, 32);
    if (lane == 0) s_red[wave] = lsum;
    __syncthreads();
    if (tid == 0) {
        float s = 0.0f;
        for (int w = 0; w < 8; ++w) s += s_red[w];
        partial[blockIdx.x] = s;
    }
}

// Fixed-order final sum: bitwise-deterministic across graph replays.
__global__ void vq_finalize(const float* __restrict__ partial, float* __restrict__ out)
{
    if (threadIdx.x == 0 && blockIdx.x == 0) {
        float s = 0.0f;
        for (int i = 0; i < 64; ++i) s += partial[i];
        float loss = s / (float)(NPOINTS * 4);   // mean over N*C elements
        out[LOSS0] = loss;    // codebook_loss
        out[LOSS1] = loss;    // commitment_loss (numerically identical)
    }
}

extern "C" void kernel_launch(void* const* d_in, const int* in_sizes, int n_in,
                              void* d_out, int out_size, void* d_ws, size_t ws_size,
                              hipStream_t stream) {
    const float* x   = (const float*)d_in[0];   // [8,4,64,64] f32
    const float* emb = (const float*)d_in[1];   // [4096,4]    f32
    float* out       = (float*)d_out;           // quant | loss | loss | indices
    float* partial   = (float*)d_ws;            // 64 block partial sums

    // 64 blocks x 8 waves x 64 points = 32768 points, exact cover (EXEC all-1s)
    vq_argmin_kernel<<<64, 256, 0, stream>>>(x, emb, out, partial);
    vq_finalize<<<1, 32, 0, stream>>>(partial, out);
}